// LaxCat_33225867001901
// MI455X (gfx1250) — compile-verified
//
#include <hip/hip_runtime.h>
#include <math.h>

namespace {
constexpr int kB     = 256;
constexpr int kT     = 3000;
constexpr int kP     = 128;
constexpr int kLout  = 999;
constexpr int kLab   = 10;
constexpr int kTiles = 63;          // ceil(999/16)
constexpr int kHJ    = 132;         // j-stride (padded) in h LDS
constexpr int kHL    = 2 * kHJ;     // 264, l-stride in h LDS
constexpr int kT2S   = 132;         // l-stride (padded) in t2 LDS
}

typedef __attribute__((ext_vector_type(2))) float v2f;
typedef __attribute__((ext_vector_type(4))) float v4f;
typedef __attribute__((ext_vector_type(8))) float v8f;

__device__ __forceinline__ float sigm(float v) { return 1.0f / (1.0f + expf(-v)); }

// ---------------------------------------------------------------------------
// Kernel 1: grouped conv (h on the fly) + variable attention per (b, 16-l tile)
// One wave handles 16 consecutive l for one b. 2 waves per block.
// Writes softmaxed A directly into d_out's attn region; outv -> workspace.
// ---------------------------------------------------------------------------
__global__ __launch_bounds__(64) void k1_conv_varattn(
    const float* __restrict__ x,       // (B, T, P)
    const float* __restrict__ conv_w,  // (P, 2, 4)
    const float* __restrict__ conv_b,  // (P, 2)
    const float* __restrict__ w1,      // (1, P)
    const float* __restrict__ b1,      // (1, 2)
    const float* __restrict__ w2,      // (2, P)
    const float* __restrict__ b2,      // (1, P)
    float* __restrict__ attn_out,      // (B, Lout, P)  [pre-battn]
    float* __restrict__ outv_ws)       // (B, Lout, 2)
{
    __shared__ float h_lds[2][16 * kHL];      // [wave][l*264 + j*132 + p]
    __shared__ float t2_lds[2][16 * kT2S];    // [wave][l*132 + p]
    __shared__ float srow[2][16], sinv[2][16];
    __shared__ float w1s[kP], w2s[2 * kP], b2s[kP], b1s[2];

    const int lane = threadIdx.x & 31;
    const int wv   = threadIdx.x >> 5;
    const int gw   = blockIdx.x * 2 + wv;     // 8064*2 == 256*63 exactly
    const int b    = gw / kTiles;
    const int l0   = (gw % kTiles) * 16;

    // Stage small parameters in LDS (shared by both waves).
    for (int i = threadIdx.x; i < kP; i += 64) {
        w1s[i]        = w1[i];
        w2s[i]        = w2[i];
        w2s[kP + i]   = w2[kP + i];
        b2s[i]        = b2[i];
    }
    if (threadIdx.x < 2) b1s[threadIdx.x] = b1[threadIdx.x];
    __syncthreads();

    // Per-lane conv weights: this lane owns p = 4*lane .. 4*lane+3.
    float cw[4][2][4];
    float cb[4][2];
    {
        const float* cwp = conv_w + lane * 32;   // 32 contiguous floats
        const float* cbp = conv_b + lane * 8;
        #pragma unroll
        for (int pp = 0; pp < 4; ++pp) {
            #pragma unroll
            for (int j = 0; j < 2; ++j) {
                #pragma unroll
                for (int k = 0; k < 4; ++k) cw[pp][j][k] = cwp[pp * 8 + j * 4 + k];
                cb[pp][j] = cbp[pp * 2 + j];
            }
        }
    }

    float* H = h_lds[wv];

    // ---- Phase A: conv -> h tile in LDS -----------------------------------
    #pragma unroll 1
    for (int l = 0; l < 16; ++l) {
        const int lt = l0 + l;                 // wave-uniform
        float acc[4][2];
        #pragma unroll
        for (int pp = 0; pp < 4; ++pp) { acc[pp][0] = 0.f; acc[pp][1] = 0.f; }
        if (lt < kLout) {
            const float* xp = x + ((size_t)b * kT + 3 * (size_t)lt) * kP + lane * 4;
            #pragma unroll
            for (int pp = 0; pp < 4; ++pp) { acc[pp][0] = cb[pp][0]; acc[pp][1] = cb[pp][1]; }
            #pragma unroll
            for (int k = 0; k < 4; ++k) {
                v4f xv = *(const v4f*)(xp + (size_t)k * kP);
                #pragma unroll
                for (int pp = 0; pp < 4; ++pp) {
                    acc[pp][0] += xv[pp] * cw[pp][0][k];
                    acc[pp][1] += xv[pp] * cw[pp][1][k];
                }
            }
        }
        #pragma unroll
        for (int j = 0; j < 2; ++j) {
            v4f hv = { acc[0][j], acc[1][j], acc[2][j], acc[3][j] };
            *(v4f*)(&H[l * kHL + j * kHJ + lane * 4]) = hv;
        }
    }
    __syncthreads();

    // ---- Phase B: t1[l][j] = sigmoid(sum_p w1[p]*h[l,p,j] + b1[j]) --------
    // lane q: l = q&15, j = q>>4  (32 lanes == 16 l * 2 j)
    float t1v;
    {
        const int lq = lane & 15, jq = lane >> 4;
        const float* hr = &H[lq * kHL + jq * kHJ];
        float s = 0.f;
        #pragma unroll 8
        for (int p = 0; p < kP; ++p) s += w1s[p] * hr[p];
        t1v = sigm(s + b1s[jq]);
    }

    // ---- Phase C: t2 = sigmoid(t1 @ w2 + b2) via V_WMMA_F32_16X16X4_F32 ---
    // A (16x4): lanes 0-15 hold K=0,1 = t1[l,0..1]; lanes 16-31 hold K=2,3 = 0
    {
        float t1_other = __shfl_xor(t1v, 16, 32);  // lane<16 receives (l, j=1)
        v2f amat;
        amat[0] = (lane < 16) ? t1v      : 0.0f;
        amat[1] = (lane < 16) ? t1_other : 0.0f;
        float* T2 = t2_lds[wv];
        const int nlo = lane & 15;
        const int mhi = (lane >> 4) * 8;
        #pragma unroll 1
        for (int c = 0; c < 8; ++c) {              // 8 chunks of 16 p
            v2f bmat;                              // B (4x16): K=0,1 rows live in lanes 0-15
            bmat[0] = (lane < 16) ? w2s[c * 16 + nlo]       : 0.0f;
            bmat[1] = (lane < 16) ? w2s[kP + c * 16 + nlo]  : 0.0f;
            v8f acc = {};
            acc = __builtin_amdgcn_wmma_f32_16x16x4_f32(
                false, amat, false, bmat, (short)0, acc, false, false);
            const int p  = c * 16 + nlo;
            const float bb = b2s[p];
            #pragma unroll
            for (int r = 0; r < 8; ++r)
                T2[(r + mhi) * kT2S + p] = sigm(acc[r] + bb);
        }
    }
    __syncthreads();

    // ---- Phase D: softmax stats over p per l (2 lanes per l) --------------
    {
        const float* T2 = t2_lds[wv];
        const int ld = lane >> 1, half = lane & 1;
        const float* row = &T2[ld * kT2S + half * 64];
        float m = -1e30f;
        #pragma unroll 8
        for (int i = 0; i < 64; ++i) m = fmaxf(m, row[i]);
        m = fmaxf(m, __shfl_xor(m, 1, 32));
        float e = 0.f;
        #pragma unroll 4
        for (int i = 0; i < 64; ++i) e += expf(row[i] - m);
        e += __shfl_xor(e, 1, 32);
        if (!half) { srow[wv][ld] = m; sinv[wv][ld] = 1.0f / e; }
    }
    __syncthreads();

    // ---- Phase E: write A (coalesced) + outv reduction --------------------
    {
        const float* T2 = t2_lds[wv];
        const size_t rowbase = ((size_t)b * kLout + l0) * kP;
        #pragma unroll 1
        for (int l = 0; l < 16; ++l) {
            const int lt = l0 + l;
            if (lt >= kLout) break;               // wave-uniform
            const float mm = srow[wv][l], iv = sinv[wv][l];
            float o0 = 0.f, o1 = 0.f;
            #pragma unroll
            for (int pc = 0; pc < 4; ++pc) {
                const int p = pc * 32 + lane;
                const float a = expf(T2[l * kT2S + p] - mm) * iv;
                attn_out[rowbase + (size_t)l * kP + p] = a;
                o0 += a * H[l * kHL + p];
                o1 += a * H[l * kHL + kHJ + p];
            }
            #pragma unroll
            for (int m = 16; m >= 1; m >>= 1) {
                o0 += __shfl_xor(o0, m, 32);
                o1 += __shfl_xor(o1, m, 32);
            }
            if (lane == 0) {
                outv_ws[((size_t)b * kLout + lt) * 2]     = o0;
                outv_ws[((size_t)b * kLout + lt) * 2 + 1] = o1;
            }
        }
    }
}

// ---------------------------------------------------------------------------
// Kernel 2: temporal attention + logits. One 256-thread block per batch b.
// ---------------------------------------------------------------------------
__global__ __launch_bounds__(256) void k2_tempattn(
    const float* __restrict__ outv_ws,  // (B, Lout, 2)
    const float* __restrict__ w1t,      // (1, Lout)
    const float* __restrict__ b1t,      // (1, 2)
    const float* __restrict__ w2t,      // (2, Lout)
    const float* __restrict__ b2t,      // (1, Lout)
    const float* __restrict__ fc_w,     // (10, 2)
    const float* __restrict__ fc_b,     // (10,)
    float* __restrict__ logits,         // (B, 10)
    float* __restrict__ battn_ws)       // (B, Lout)
{
    __shared__ float red[256];
    __shared__ float s2buf[kLout];
    __shared__ float sh[4];

    const int b = blockIdx.x, tid = threadIdx.x;
    const float* ov = outv_ws + (size_t)b * kLout * 2;

    // s1 = sigmoid(sum_l w1t[l]*outv[b,l,:] + b1t)
    float a0 = 0.f, a1 = 0.f;
    for (int l = tid; l < kLout; l += 256) {
        const float w = w1t[l];
        a0 += w * ov[l * 2];
        a1 += w * ov[l * 2 + 1];
    }
    red[tid] = a0; __syncthreads();
    for (int s = 128; s > 0; s >>= 1) { if (tid < s) red[tid] += red[tid + s]; __syncthreads(); }
    if (tid == 0) sh[0] = red[0];
    __syncthreads();
    red[tid] = a1; __syncthreads();
    for (int s = 128; s > 0; s >>= 1) { if (tid < s) red[tid] += red[tid + s]; __syncthreads(); }
    if (tid == 0) sh[1] = red[0];
    __syncthreads();

    const float s10 = sigm(sh[0] + b1t[0]);
    const float s11 = sigm(sh[1] + b1t[1]);
    const float scale = rsqrtf((float)kLout);

    // s2 (pre-scaled) + running max
    float lm = -1e30f;
    for (int l = tid; l < kLout; l += 256) {
        const float v = sigm(s10 * w2t[l] + s11 * w2t[kLout + l] + b2t[l]) * scale;
        s2buf[l] = v;
        lm = fmaxf(lm, v);
    }
    red[tid] = lm; __syncthreads();
    for (int s = 128; s > 0; s >>= 1) { if (tid < s) red[tid] = fmaxf(red[tid], red[tid + s]); __syncthreads(); }
    if (tid == 0) sh[2] = red[0];
    __syncthreads();
    const float mx = sh[2];

    float es = 0.f;
    for (int l = tid; l < kLout; l += 256) es += expf(s2buf[l] - mx);
    red[tid] = es; __syncthreads();
    for (int s = 128; s > 0; s >>= 1) { if (tid < s) red[tid] += red[tid + s]; __syncthreads(); }
    if (tid == 0) sh[3] = red[0];
    __syncthreads();
    const float inv = 1.0f / sh[3];

    float o0 = 0.f, o1 = 0.f;
    for (int l = tid; l < kLout; l += 256) {
        const float bv = expf(s2buf[l] - mx) * inv;
        battn_ws[(size_t)b * kLout + l] = bv;
        o0 += bv * ov[l * 2];
        o1 += bv * ov[l * 2 + 1];
    }
    red[tid] = o0; __syncthreads();
    for (int s = 128; s > 0; s >>= 1) { if (tid < s) red[tid] += red[tid + s]; __syncthreads(); }
    if (tid == 0) sh[0] = red[0];
    __syncthreads();
    red[tid] = o1; __syncthreads();
    for (int s = 128; s > 0; s >>= 1) { if (tid < s) red[tid] += red[tid + s]; __syncthreads(); }
    if (tid == 0) sh[1] = red[0];
    __syncthreads();

    if (tid < kLab)
        logits[(size_t)b * kLab + tid] =
            sh[0] * fc_w[tid * 2] + sh[1] * fc_w[tid * 2 + 1] + fc_b[tid];
}

// ---------------------------------------------------------------------------
// Kernel 3: attn[b,l,p] *= battn[b,l]  (vectorized float4, in place in d_out)
// ---------------------------------------------------------------------------
__global__ __launch_bounds__(256) void k3_scale(
    float* __restrict__ attn, const float* __restrict__ battn, size_t n4)
{
    const size_t i = (size_t)blockIdx.x * 256 + threadIdx.x;
    if (i < n4) {
        v4f* p = (v4f*)attn + i;
        v4f v = *p;
        v *= battn[i >> 5];         // 32 float4 per (b,l) row of 128 p
        *p = v;
    }
}

extern "C" void kernel_launch(void* const* d_in, const int* in_sizes, int n_in,
                              void* d_out, int out_size, void* d_ws, size_t ws_size,
                              hipStream_t stream) {
    const float* x      = (const float*)d_in[0];
    const float* conv_w = (const float*)d_in[1];
    const float* conv_b = (const float*)d_in[2];
    const float* w1     = (const float*)d_in[3];
    const float* b1     = (const float*)d_in[4];
    const float* w2     = (const float*)d_in[5];
    const float* b2     = (const float*)d_in[6];
    const float* w1t    = (const float*)d_in[7];
    const float* b1t    = (const float*)d_in[8];
    const float* w2t    = (const float*)d_in[9];
    const float* b2t    = (const float*)d_in[10];
    const float* fc_w   = (const float*)d_in[11];
    const float* fc_b   = (const float*)d_in[12];

    float* logits = (float*)d_out;                 // (256, 10)
    float* attn   = logits + (size_t)kB * kLab;    // (256, 999, 128)

    float* outv_ws  = (float*)d_ws;                          // (256, 999, 2)
    float* battn_ws = outv_ws + (size_t)kB * kLout * 2;      // (256, 999)

    // 256 b * 63 l-tiles = 16128 waves, 2 waves (64 thr) per block
    k1_conv_varattn<<<dim3(8064), dim3(64), 0, stream>>>(
        x, conv_w, conv_b, w1, b1, w2, b2, attn, outv_ws);

    k2_tempattn<<<dim3(kB), dim3(256), 0, stream>>>(
        outv_ws, w1t, b1t, w2t, b2t, fc_w, fc_b, logits, battn_ws);

    const size_t n4 = (size_t)kB * kLout * (kP / 4);
    k3_scale<<<dim3((unsigned)((n4 + 255) / 256)), dim3(256), 0, stream>>>(
        attn, battn_ws, n4);
}